// DeformableConv2d_22204980920787
// MI455X (gfx1250) — compile-verified
//
#include <hip/hip_runtime.h>
#include <hip/hip_bf16.h>

typedef __attribute__((ext_vector_type(2))) float v2f;
typedef __attribute__((ext_vector_type(8))) float v8f;

#define C_IN 256
#define COUT 256
#define HW   4096     // 64*64
#define CK   2304     // C_IN*9
#define MT   128      // M tile per workgroup
#define NT   64       // N tile per workgroup
#define CT   32       // c-chunk (K-depth per stage)
#define LP   (CT + 2) // padded LDS row stride (even -> 8B-aligned v2f frags)

#if __has_builtin(__builtin_amdgcn_global_load_async_to_lds_b32) && \
    __has_builtin(__builtin_amdgcn_s_wait_asynccnt)
#define USE_ASYNC_COPY 1
typedef __attribute__((address_space(1))) int g_int;   // global (AS1)
typedef __attribute__((address_space(3))) int s_int;   // LDS (AS3)
#else
#define USE_ASYNC_COPY 0
#endif

// ---------------------------------------------------------------------------
// Stage 1: offset (18ch) + mask (9ch) 3x3 convs, then coords py/px and
// sigmoid(mask), stored as [B,9,H,W] float arrays in workspace.
// One 64-thread block (2 waves) per output pixel.
// ---------------------------------------------------------------------------
__global__ __launch_bounds__(64) void offmask_kernel(
    const float* __restrict__ x,
    const float* __restrict__ w_off,  const float* __restrict__ b_off,
    const float* __restrict__ w_mask, const float* __restrict__ b_mask,
    float* __restrict__ pyA, float* __restrict__ pxA, float* __restrict__ mA)
{
    const int p   = blockIdx.x;          // 0 .. B*HW-1
    const int b   = p >> 12;
    const int hw  = p & 4095;
    const int h   = hw >> 6;
    const int w   = hw & 63;
    const int tid = threadIdx.x;

    float acc[27];
#pragma unroll
    for (int i = 0; i < 27; ++i) acc[i] = 0.f;

    const float* xb = x + (size_t)b * C_IN * HW;

    for (int cc = 0; cc < 4; ++cc) {
        const int c = tid + cc * 64;
        const float* xc = xb + (size_t)c * HW;
#pragma unroll
        for (int t = 0; t < 9; ++t) {
            const int ky = t / 3, kx = t % 3;
            const int yy = h - 1 + ky, xx = w - 1 + kx;
            const float xv = (yy >= 0 && yy < 64 && xx >= 0 && xx < 64)
                                 ? xc[yy * 64 + xx] : 0.f;
            const int wbase = c * 9 + t;
#pragma unroll
            for (int oc = 0; oc < 18; ++oc)
                acc[oc] += xv * w_off[(size_t)oc * CK + wbase];
#pragma unroll
            for (int oc = 0; oc < 9; ++oc)
                acc[18 + oc] += xv * w_mask[(size_t)oc * CK + wbase];
        }
    }

    __shared__ float red[27][65];
    __shared__ float sums[27];
#pragma unroll
    for (int i = 0; i < 27; ++i) red[i][tid] = acc[i];
    __syncthreads();
    if (tid < 27) {
        float s = 0.f;
        for (int l = 0; l < 64; ++l) s += red[tid][l];
        sums[tid] = s;
    }
    __syncthreads();
    if (tid < 9) {
        const float dy = sums[2 * tid]     + b_off[2 * tid];
        const float dx = sums[2 * tid + 1] + b_off[2 * tid + 1];
        float mm = sums[18 + tid] + b_mask[tid];
        mm = 1.f / (1.f + __expf(-mm));
        const float py = (float)(h - 1 + tid / 3) + dy;
        const float px = (float)(w - 1 + tid % 3) + dx;
        const size_t o = ((size_t)b * 9 + tid) * HW + hw;
        pyA[o] = py; pxA[o] = px; mA[o] = mm;
    }
}

// ---------------------------------------------------------------------------
// Stage 2+3 fused: bilinear sample into LDS B-tiles, weights into LDS A-tiles
// (async global->LDS when available), then fp32 WMMA GEMM
//   out[M=256, N=16384] = W[256, 2304] x val[2304, 16384].
// Grid (N/NT, M/MT), 256 threads = 8 waves; wave w owns a 16-row M slab x
// four 16-col N subtiles (4 x v8f accumulators).
// Both A and B tiles are stored K-fastest with even row stride so each WMMA
// fragment is one aligned ds_load_b64 into an even VGPR pair.
// ---------------------------------------------------------------------------
__global__ __launch_bounds__(256) void dcn_gemm_kernel(
    const float* __restrict__ x,
    const float* __restrict__ w_dcn, const float* __restrict__ b_dcn,
    const float* __restrict__ pyA, const float* __restrict__ pxA,
    const float* __restrict__ mA,
    float* __restrict__ out)
{
    __shared__ float ldsA[MT][LP];    // [m][c], stride 34 (even)
    __shared__ float ldsBt[NT][LP];   // [n][c], transposed, stride 34
    __shared__ int   mY0[9][NT], mY1[9][NT], mX0[9][NT], mX1[9][NT];
    __shared__ float mW00[9][NT], mW01[9][NT], mW10[9][NT], mW11[9][NT];

    const int tid = threadIdx.x;
    const int n0  = blockIdx.x * NT;
    const int m0  = blockIdx.y * MT;

    // Precompute bilinear metadata for all 9*NT (k, n) pairs of this tile.
    for (int idx = tid; idx < 9 * NT; idx += 256) {
        const int kk = idx >> 6;
        const int nl = idx & (NT - 1);
        const int n  = n0 + nl;
        const int b  = n >> 12;
        const int hw = n & 4095;
        const size_t o = ((size_t)b * 9 + kk) * HW + hw;
        const float py = pyA[o], px = pxA[o], mm = mA[o];
        const float fy = floorf(py), fx = floorf(px);
        const int y0 = (int)fy, x0 = (int)fx;
        const float ty = py - fy, tx = px - fx;
        const bool vy0 = (y0 >= 0) & (y0 <= 63);
        const bool vy1 = (y0 + 1 >= 0) & (y0 + 1 <= 63);
        const bool vx0 = (x0 >= 0) & (x0 <= 63);
        const bool vx1 = (x0 + 1 >= 0) & (x0 + 1 <= 63);
        mW00[kk][nl] = (vy0 && vx0) ? (1.f - ty) * (1.f - tx) * mm : 0.f;
        mW01[kk][nl] = (vy0 && vx1) ? (1.f - ty) * tx * mm : 0.f;
        mW10[kk][nl] = (vy1 && vx0) ? ty * (1.f - tx) * mm : 0.f;
        mW11[kk][nl] = (vy1 && vx1) ? ty * tx * mm : 0.f;
        mY0[kk][nl] = min(max(y0, 0), 63);
        mY1[kk][nl] = min(max(y0 + 1, 0), 63);
        mX0[kk][nl] = min(max(x0, 0), 63);
        mX1[kk][nl] = min(max(x0 + 1, 0), 63);
    }

    const int wave  = tid >> 5;
    const int lane  = tid & 31;
    const int lm    = lane & 15;
    const int khalf = (lane >> 4) << 1;   // 0 or 2 (f32 WMMA A/B lane split)

    v8f acc[4];
#pragma unroll
    for (int ns = 0; ns < 4; ++ns)
        acc[ns] = (v8f){0.f, 0.f, 0.f, 0.f, 0.f, 0.f, 0.f, 0.f};

    for (int kk = 0; kk < 9; ++kk) {
        for (int c0 = 0; c0 < C_IN; c0 += CT) {
            __syncthreads();
            // --- stage A tile: weights w_dcn[m0+m][(c0+c)*9 + kk]
#if USE_ASYNC_COPY
            for (int idx = tid; idx < MT * CT; idx += 256) {
                const int m = idx >> 5;         // /CT
                const int c = idx & (CT - 1);
                float* gsrc = const_cast<float*>(
                    &w_dcn[(size_t)(m0 + m) * CK + (size_t)(c0 + c) * 9 + kk]);
                __builtin_amdgcn_global_load_async_to_lds_b32(
                    (g_int*)gsrc, (s_int*)&ldsA[m][c], 0, 0);
            }
            __builtin_amdgcn_s_wait_asynccnt(0);
#else
            for (int idx = tid; idx < MT * CT; idx += 256) {
                const int m = idx >> 5;         // /CT
                const int c = idx & (CT - 1);
                ldsA[m][c] = w_dcn[(size_t)(m0 + m) * CK + (size_t)(c0 + c) * 9 + kk];
                if (c0 + CT < C_IN)             // prefetch next weight chunk
                    __builtin_prefetch(
                        &w_dcn[(size_t)(m0 + m) * CK + (size_t)(c0 + CT + c) * 9 + kk], 0, 3);
            }
#endif
            // --- stage B tile (transposed): bilinear-sampled, masked values
            for (int idx = tid; idx < CT * NT; idx += 256) {
                const int c  = idx >> 6;
                const int nl = idx & (NT - 1);
                const int n  = n0 + nl;
                const float* xc = x + (((size_t)(n >> 12)) * C_IN + (c0 + c)) * HW;
                const int y0 = mY0[kk][nl], y1 = mY1[kk][nl];
                const int x0 = mX0[kk][nl], x1 = mX1[kk][nl];
                ldsBt[nl][c] = mW00[kk][nl] * xc[y0 * 64 + x0]
                             + mW01[kk][nl] * xc[y0 * 64 + x1]
                             + mW10[kk][nl] * xc[y1 * 64 + x0]
                             + mW11[kk][nl] * xc[y1 * 64 + x1];
            }
            __syncthreads();
            // --- fp32 WMMA over the 32-deep chunk (8 steps of K=4)
#pragma unroll
            for (int k4 = 0; k4 < CT / 4; ++k4) {
                const int kb = k4 * 4 + khalf;  // always even
                const v2f a =
                    *reinterpret_cast<const v2f*>(&ldsA[(wave << 4) + lm][kb]);
#pragma unroll
                for (int ns = 0; ns < 4; ++ns) {
                    const v2f bb =
                        *reinterpret_cast<const v2f*>(&ldsBt[(ns << 4) + lm][kb]);
                    acc[ns] = __builtin_amdgcn_wmma_f32_16x16x4_f32(
                        false, a, false, bb, (short)0, acc[ns], false, false);
                }
            }
        }
    }

    // --- epilogue: C/D layout — VGPR r: M = r + 8*(lane>=16), N = lane%16
    const int mrowbase = m0 + (wave << 4) + ((lane >> 4) << 3);
#pragma unroll
    for (int ns = 0; ns < 4; ++ns) {
        const int n  = n0 + (ns << 4) + lm;
        const int b  = n >> 12;
        const int hw = n & 4095;
#pragma unroll
        for (int r = 0; r < 8; ++r) {
            const int oc = mrowbase + r;
            out[(((size_t)b * COUT) + oc) * HW + hw] = acc[ns][r] + b_dcn[oc];
        }
    }
}

// ---------------------------------------------------------------------------
extern "C" void kernel_launch(void* const* d_in, const int* in_sizes, int n_in,
                              void* d_out, int out_size, void* d_ws, size_t ws_size,
                              hipStream_t stream) {
    const float* x      = (const float*)d_in[0];
    const float* w_off  = (const float*)d_in[1];
    const float* b_off  = (const float*)d_in[2];
    const float* w_mask = (const float*)d_in[3];
    const float* b_mask = (const float*)d_in[4];
    const float* w_dcn  = (const float*)d_in[5];
    const float* b_dcn  = (const float*)d_in[6];
    float* outp = (float*)d_out;

    // workspace: py / px / mask arrays, each [B=4][K=9][HW=4096] floats
    float* pyA = (float*)d_ws;
    float* pxA = pyA + (size_t)4 * 9 * HW;
    float* mA  = pxA + (size_t)4 * 9 * HW;

    offmask_kernel<<<dim3(4 * HW), dim3(64), 0, stream>>>(
        x, w_off, b_off, w_mask, b_mask, pyA, pxA, mA);

    dcn_gemm_kernel<<<dim3((4 * HW) / NT, COUT / MT), dim3(256), 0, stream>>>(
        x, w_dcn, b_dcn, pyA, pxA, mA, outp);
}